// RepLKBlock_1151051235432
// MI455X (gfx1250) — compile-verified
//
#include <hip/hip_runtime.h>

typedef __attribute__((ext_vector_type(2))) float v2f;
typedef __attribute__((ext_vector_type(8))) float v8f;

#define BN_EPS   1e-5f
#define C_CH     384
#define IMG      64
#define WSTRIDE  170      // 169 merged weights + 1 bias per channel
#define XW       44       // 32 + 2*6 halo
#define NCHUNK   91       // 13 kernel rows * 7 K-chunks of 4

// ---------------------------------------------------------------------------
// Prep: fold BN into weights, embed 3x3 into center of 13x13, compute bias.
// wm layout per channel: [0..168] merged 13x13 weights, [169] bias.
// ---------------------------------------------------------------------------
__global__ __launch_bounds__(256) void replk_merge_weights(
    const float* __restrict__ lw, const float* __restrict__ lg,
    const float* __restrict__ lb, const float* __restrict__ lm,
    const float* __restrict__ lv,
    const float* __restrict__ sw, const float* __restrict__ sg,
    const float* __restrict__ sb, const float* __restrict__ sm,
    const float* __restrict__ sv,
    float* __restrict__ wm)
{
    int idx = blockIdx.x * blockDim.x + threadIdx.x;
    if (idx >= C_CH * WSTRIDE) return;
    int c = idx / WSTRIDE;
    int i = idx - c * WSTRIDE;
    float scale_l = lg[c] / sqrtf(lv[c] + BN_EPS);
    float scale_s = sg[c] / sqrtf(sv[c] + BN_EPS);
    if (i == 169) {
        wm[idx] = (lb[c] - lm[c] * scale_l) + (sb[c] - sm[c] * scale_s);
    } else {
        int kr = i / 13, kc = i - kr * 13;
        float v = lw[c * 169 + i] * scale_l;
        if (kr >= 5 && kr < 8 && kc >= 5 && kc < 8)
            v += sw[c * 9 + (kr - 5) * 3 + (kc - 5)] * scale_s;
        wm[idx] = v;
    }
}

// ---------------------------------------------------------------------------
// Main: 13x13 depthwise conv + bias via V_WMMA_F32_16X16X4_F32.
// Block = 128 threads (4 waves). Block computes a 32x32 output tile of one
// (b,c) image; each wave owns a 16x16 quadrant and issues 13*7 = 91 WMMAs.
// ---------------------------------------------------------------------------
__global__ __launch_bounds__(128) void replk_dwconv13_wmma(
    const float* __restrict__ x, const float* __restrict__ wm,
    float* __restrict__ out)
{
    __shared__ float Xs[XW * XW];          // 44x44 haloed input tile (7.7 KB)
    __shared__ float Bs[NCHUNK * 64];      // banded B matrices, WMMA lane layout (23.3 KB)
    __shared__ float Ws[WSTRIDE];          // merged weights + bias

    const int tid = threadIdx.x;
    int t = blockIdx.x;
    const int quad = t & 3;  t >>= 2;
    const int c = t % C_CH;
    const int b = t / C_CH;
    const int row0 = (quad >> 1) * 32;
    const int col0 = (quad & 1) * 32;

    // Stage merged weights for this channel.
    const float* wc = wm + c * WSTRIDE;
    for (int i = tid; i < WSTRIDE; i += 128) Ws[i] = wc[i];
    __syncthreads();

    // Expand banded B matrices into WMMA B layout:
    //   chunk (kr,q): lane l holds B[K=2h+v][n] = W[kr][4q+2h+v-n], h=l/16, n=l%16.
    //   Stored so each lane fetches its 2 VGPRs with one b64 load: e = chunk*64 + l*2 + v.
    for (int e = tid; e < NCHUNK * 64; e += 128) {
        int chunk = e >> 6;
        int lane2 = e & 63;
        int lane  = lane2 >> 1;
        int v     = lane2 & 1;
        int kr = chunk / 7, q = chunk - kr * 7;
        int h = lane >> 4, n = lane & 15;
        int tap = 4 * q + 2 * h + v - n;
        Bs[e] = (tap >= 0 && tap < 13) ? Ws[kr * 13 + tap] : 0.0f;
    }

    // Stage 44x44 haloed input tile (zero padding at image borders).
    const float* xc = x + (long)(b * C_CH + c) * (IMG * IMG);
    for (int e = tid; e < XW * XW; e += 128) {
        int rr = e / XW, cc = e - rr * XW;
        int gr = row0 - 6 + rr;
        int gc = col0 - 6 + cc;
        float val = 0.0f;
        if (gr >= 0 && gr < IMG && gc >= 0 && gc < IMG)
            val = xc[gr * IMG + gc];
        Xs[e] = val;
    }
    __syncthreads();

    const int wave = tid >> 5;
    const int lane = tid & 31;
    const int qr = (wave >> 1) * 16;   // quadrant origin within 32x32 tile
    const int qc = (wave & 1) * 16;
    const int m = lane & 15;           // output row within quadrant (A-matrix M)
    const int h = lane >> 4;           // lane half selects K=0/1 vs K=2/3

    const float bias = Ws[169];
    v8f acc0 = {bias, bias, bias, bias, bias, bias, bias, bias};
    v8f acc1 = {0.f, 0.f, 0.f, 0.f, 0.f, 0.f, 0.f, 0.f};

    for (int kr = 0; kr < 13; ++kr) {
        // A chunk q: lane reads X[qr+m+kr][qc + 4q + 2h .. +1]  (8B aligned, bank-clean)
        const float* arow = &Xs[(qr + m + kr) * XW + qc + 2 * h];
        const float* brow = &Bs[kr * 7 * 64 + lane * 2];
        #pragma unroll
        for (int q = 0; q < 7; ++q) {
            v2f A = *(const v2f*)(arow + 4 * q);
            v2f B = *(const v2f*)(brow + q * 64);
            if (kr & 1)
                acc1 = __builtin_amdgcn_wmma_f32_16x16x4_f32(
                           false, A, false, B, (short)0, acc1, false, false);
            else
                acc0 = __builtin_amdgcn_wmma_f32_16x16x4_f32(
                           false, A, false, B, (short)0, acc0, false, false);
        }
    }
    v8f acc = acc0 + acc1;

    // D layout: VGPR v -> row (qr + v + 8h), col = qc + (lane%16).
    float* oc = out + (long)(b * C_CH + c) * (IMG * IMG);
    const int orow = row0 + qr + 8 * h;
    const int ocol = col0 + qc + m;
    #pragma unroll
    for (int v = 0; v < 8; ++v)
        oc[(orow + v) * IMG + ocol] = acc[v];
}

// ---------------------------------------------------------------------------
extern "C" void kernel_launch(void* const* d_in, const int* in_sizes, int n_in,
                              void* d_out, int out_size, void* d_ws, size_t ws_size,
                              hipStream_t stream) {
    const float* x  = (const float*)d_in[0];
    const float* lw = (const float*)d_in[1];
    const float* lg = (const float*)d_in[2];
    const float* lb = (const float*)d_in[3];
    const float* lm = (const float*)d_in[4];
    const float* lv = (const float*)d_in[5];
    const float* sw = (const float*)d_in[6];
    const float* sg = (const float*)d_in[7];
    const float* sb = (const float*)d_in[8];
    const float* sm = (const float*)d_in[9];
    const float* sv = (const float*)d_in[10];
    float* outp = (float*)d_out;
    float* wmerged = (float*)d_ws;   // C_CH * WSTRIDE floats (~261 KB)

    int prep_n = C_CH * WSTRIDE;
    replk_merge_weights<<<(prep_n + 255) / 256, 256, 0, stream>>>(
        lw, lg, lb, lm, lv, sw, sg, sb, sm, sv, wmerged);

    int B = 16;
    int nblocks = B * C_CH * 4;      // 4 quadrant-blocks of 32x32 per 64x64 image
    replk_dwconv13_wmma<<<nblocks, 128, 0, stream>>>(x, wmerged, outp);
}